// FLAttention_80839874445788
// MI455X (gfx1250) — compile-verified
//
#include <hip/hip_runtime.h>
#include <hip/hip_bf16.h>

typedef __attribute__((ext_vector_type(2))) float v2f;
typedef __attribute__((ext_vector_type(8))) float v8f;

#define DD   1024
#define BB   16
#define HH   4
#define EPSF 1e-8f
#define LOG2E 1.44269504088896340736f

// One block: fixed batch b, 16 query rows, 4 waves (wave w == head w).
// Per wave: two-pass softmax over j=0..1023; attend-accumulate via
// V_WMMA_F32_16X16X4_F32 (A = P tile 16x4, B = v broadcast 4x16).
__global__ __launch_bounds__(128) void flatt_kernel(
    const float* __restrict__ x,
    const float* __restrict__ aq, const float* __restrict__ bq,
    const float* __restrict__ ak, const float* __restrict__ bk,
    const float* __restrict__ av, const float* __restrict__ bv,
    float* __restrict__ out)
{
    __shared__ float xs[DD];        // x[b, :]
    __shared__ float att_s[HH * 16];
    __shared__ float dn_s[HH * 16];

    const int b     = blockIdx.x >> 6;   // 64 i-tiles per batch row
    const int itile = blockIdx.x & 63;
    const int i0    = itile << 4;
    const int tid   = threadIdx.x;
    const int h     = tid >> 5;          // wave id == head index (wave32)
    const int lane  = tid & 31;
    const int iloc  = lane & 15;         // matrix row M this lane owns
    const int half  = lane >> 4;         // K-half: lanes 0-15 -> K=0,1 ; 16-31 -> K=2,3

    // Stage x[b,:] (4KB) into LDS, coalesced: 128 threads x 8 floats.
    #pragma unroll
    for (int r = 0; r < DD / 128; ++r)
        xs[tid + 128 * r] = x[b * DD + tid + 128 * r];
    __syncthreads();

    const float Aq = aq[h], Bq = bq[h];
    const float Ak = ak[h], Bk = bk[h];
    const float Av = av[h], Bv = bv[h];

    const float q = fmaf(xs[i0 + iloc], Aq, Bq);

    // ---- Pass 1: row max of s_ij = 1/(|k_j - q_i| + eps).
    // Lane halves partition j: half0 -> {4c,4c+1}, half1 -> {4c+2,4c+3}.
    float m = -3.4e38f;
    for (int c = 0; c < DD / 4; ++c) {
        const int j = 4 * c + 2 * half;
        const float x0 = xs[j], x1 = xs[j + 1];
        const float s0 = __builtin_amdgcn_rcpf(fabsf(fmaf(x0, Ak, Bk) - q) + EPSF);
        const float s1 = __builtin_amdgcn_rcpf(fabsf(fmaf(x1, Ak, Bk) - q) + EPSF);
        m = fmaxf(m, fmaxf(s0, s1));
    }
    // Combine with the lane handling the other K-half of the same row.
    m = fmaxf(m, __shfl_xor(m, 16, 32));

    // ---- Pass 2: P = exp(s - m); numerator via WMMA, denominator scalar.
    v8f acc = {};          // C/D: 16x16 f32, all 16 columns identical (= att_i)
    float denom = 0.0f;
    for (int c = 0; c < DD / 4; ++c) {
        const int j = 4 * c + 2 * half;
        const float x0 = xs[j], x1 = xs[j + 1];
        const float s0 = __builtin_amdgcn_rcpf(fabsf(fmaf(x0, Ak, Bk) - q) + EPSF);
        const float s1 = __builtin_amdgcn_rcpf(fabsf(fmaf(x1, Ak, Bk) - q) + EPSF);
        const float p0 = __builtin_amdgcn_exp2f((s0 - m) * LOG2E);
        const float p1 = __builtin_amdgcn_exp2f((s1 - m) * LOG2E);
        denom += p0 + p1;

        v2f A;  A.x  = p0;                  A.y  = p1;                  // A[i, K], K = 2*half+{0,1}
        v2f Bm; Bm.x = fmaf(x0, Av, Bv);    Bm.y = fmaf(x1, Av, Bv);    // B[K, *] = v_j (col-invariant)

        // D = A x B + C ; co-executes with the VALU exp/rcp stream.
        acc = __builtin_amdgcn_wmma_f32_16x16x4_f32(
                  false, A, false, Bm, (short)0, acc, false, false);
    }
    denom += __shfl_xor(denom, 16, 32);

    // C layout: VGPR r -> lanes 0-15: M=r ; lanes 16-31: M=r+8 (all N equal).
    if (lane < 16) dn_s[h * 16 + iloc] = denom;
    if (lane == 0) {
        #pragma unroll
        for (int r = 0; r < 8; ++r) att_s[h * 16 + r] = acc[r];
    }
    if (lane == 16) {
        #pragma unroll
        for (int r = 0; r < 8; ++r) att_s[h * 16 + 8 + r] = acc[r];
    }
    __syncthreads();

    // Head sum + 1/sqrt(H) + residual.
    if (tid < 16) {
        float s = 0.0f;
        #pragma unroll
        for (int hh = 0; hh < HH; ++hh)
            s += att_s[hh * 16 + tid] / dn_s[hh * 16 + tid];
        out[b * DD + i0 + tid] = xs[i0 + tid] + 0.5f * s;
    }
}

extern "C" void kernel_launch(void* const* d_in, const int* in_sizes, int n_in,
                              void* d_out, int out_size, void* d_ws, size_t ws_size,
                              hipStream_t stream) {
    (void)in_sizes; (void)n_in; (void)d_ws; (void)ws_size; (void)out_size;
    const float* x  = (const float*)d_in[0];
    const float* aq = (const float*)d_in[1];
    const float* bq = (const float*)d_in[2];
    const float* ak = (const float*)d_in[3];
    const float* bk = (const float*)d_in[4];
    const float* av = (const float*)d_in[5];
    const float* bv = (const float*)d_in[6];
    float* out = (float*)d_out;

    // 16 batches x 64 i-tiles = 1024 blocks; 128 threads = 4 wave32 (one per head).
    flatt_kernel<<<dim3(BB * (DD / 16)), dim3(128), 0, stream>>>(
        x, aq, bq, ak, bk, av, bv, out);
}